// fast_attn_76656576299161
// MI455X (gfx1250) — compile-verified
//
#include <hip/hip_runtime.h>
#include <math.h>

// Problem constants (from reference setup_inputs)
constexpr int B   = 4;
constexpr int N   = 4096;
constexpr int KNB = 16;    // neighbors
constexpr int C   = 128;   // dim
constexpr int HID = 64;
constexpr int SEG = 8;     // n-segments for kv partial sums (512 n each)
constexpr float EPS = 1e-12f;

typedef __attribute__((ext_vector_type(2))) float v2f;
typedef __attribute__((ext_vector_type(8))) float v8f;

// D(16x16) = A(16x4) * B(4x16) + C, fp32 WMMA (CDNA5 V_WMMA_F32_16X16X4_F32)
static __device__ __forceinline__ v8f wmma4(v2f a, v2f b, v8f c) {
  return __builtin_amdgcn_wmma_f32_16x16x4_f32(
      /*neg_a=*/false, a, /*neg_b=*/false, b,
      /*c_mod=*/(short)0, c, /*reuse_a=*/false, /*reuse_b=*/false);
}

// ---------------------------------------------------------------------------
// Tensor Data Mover: 2D tile global->LDS with optional per-row LDS padding.
// D# layout per CDNA5 ISA (groups 2/3 zero = 2D).
// pad codes: interval v -> 2^(v+1) dwords before pad; amount v -> v+1 dwords.
// This toolchain (clang-23 / therock) declares the 6-arg builtin:
//   (uint32x4 g0, int32x8 g1, int32x4, int32x4, int32x8, i32 cpol)
// ---------------------------------------------------------------------------
#if defined(__has_builtin)
#if __has_builtin(__builtin_amdgcn_tensor_load_to_lds)
#define HAVE_TDM 1
#endif
#endif

#ifdef HAVE_TDM
typedef unsigned int v4u __attribute__((ext_vector_type(4)));
typedef int v8i __attribute__((ext_vector_type(8)));
typedef int v4i __attribute__((ext_vector_type(4)));

static __device__ __forceinline__ void tdm_load_2d(
    const void* lds_dst, const void* gsrc,
    unsigned d0_elems,            // row length (elements, contiguous)
    unsigned d1_rows,             // number of rows
    unsigned long long row_stride_elems,
    unsigned pad_interval_code, unsigned pad_amount_code, bool pad_en) {
  const unsigned long long ga = (unsigned long long)(uintptr_t)gsrc;
  const unsigned lds_off = (unsigned)(uintptr_t)lds_dst;  // low 32 = LDS offset
  v4u g0;
  g0.x = 0x1u;                                            // count=1, user mode
  g0.y = lds_off;                                         // lds_addr
  g0.z = (unsigned)(ga & 0xFFFFFFFFull);                  // global_addr[31:0]
  g0.w = (unsigned)((ga >> 32) & 0x1FFFFFFull) | (2u << 30);  // addr[56:32]|type=2
  v8i g1;
  unsigned e0 = (2u << 16);                               // data_size = 4B
  if (pad_en) e0 |= (1u << 20) | (pad_interval_code << 22) | (pad_amount_code << 25);
  g1[0] = (int)e0;
  g1[1] = (int)((d0_elems & 0xFFFFu) << 16);              // tensor_dim0[15:0]
  g1[2] = (int)((d0_elems >> 16) | ((d1_rows & 0xFFFFu) << 16));
  g1[3] = (int)((d1_rows >> 16) | (d0_elems << 16));      // tile_dim0
  g1[4] = (int)(d1_rows & 0xFFFFu);                       // tile_dim1 (tile_dim2=0)
  g1[5] = (int)(unsigned)(row_stride_elems & 0xFFFFFFFFull);
  g1[6] = (int)(unsigned)((row_stride_elems >> 32) & 0xFFFFull);
  g1[7] = 0;
  v4i gz4 = {0, 0, 0, 0};
  v8i gz8 = {0, 0, 0, 0, 0, 0, 0, 0};
  __builtin_amdgcn_tensor_load_to_lds(g0, g1, gz4, gz4, gz8, 0);
}
#endif

// ---------------------------------------------------------------------------
// Kernel 1: qkv = xt @ W_qkv^T  -> Q,K,V [B,N,C]
// LDS layouts chosen in natural memory order so TDM can stage them:
//   XT[c][n] (stride 136, "two-row" operand reads), WL[j][c] (stride 132).
// ---------------------------------------------------------------------------
__global__ __launch_bounds__(256) void k_qkv(const float* __restrict__ x,
                                             const float* __restrict__ Wqkv,
                                             float* __restrict__ Q,
                                             float* __restrict__ Kf,
                                             float* __restrict__ V) {
  __shared__ float XT[128][136];   // x chunk   [c][n_local]
  __shared__ float WL[384][132];   // W_qkv     [j][c]
  const int b   = blockIdx.y;
  const int n0  = blockIdx.x * 128;
  const int tid = threadIdx.x;
  const int wv  = tid >> 5, lane = tid & 31, l = lane & 15, hf = lane >> 4;

#ifdef HAVE_TDM
  if (wv == 0) {
    tdm_load_2d(&XT[0][0], x + ((size_t)b * C) * N + n0, 128, 128,
                (unsigned long long)N, /*interval 128dw*/6, /*pad 8dw*/7, true);
    tdm_load_2d(&WL[0][0], Wqkv, 128, 384,
                128ull, /*interval 128dw*/6, /*pad 4dw*/3, true);
    __builtin_amdgcn_s_wait_tensorcnt(0);
  }
#else
  for (int p = 0; p < 64; ++p) {            // XT[c][nl] = x[b,c,n0+nl]
    int w = tid + p * 256;
    int c = w >> 7, nl = w & 127;
    XT[c][nl] = x[((size_t)b * C + c) * N + n0 + nl];
  }
  for (int p = 0; p < 192; ++p) {           // WL[j][c] = W_qkv[j,c]
    int w = tid + p * 256;
    int c = w & 127, j = w >> 7;
    WL[j][c] = Wqkv[j * C + c];
  }
#endif
  __syncthreads();

  for (int t = wv; t < 8 * 24; t += 8) {    // 8 m-tiles x 24 j-tiles
    const int mt = t & 7, jt = t >> 3;
    const int m0 = mt * 16;
    v8f acc = {};
    for (int c0 = 0; c0 < C; c0 += 4) {
      v2f a, bb;
      a.x  = XT[c0 + 2 * hf][m0 + l];       // A[n][c] = XT^T
      a.y  = XT[c0 + 2 * hf + 1][m0 + l];
      bb.x = WL[jt * 16 + l][c0 + 2 * hf];  // B[c][j] = W^T
      bb.y = WL[jt * 16 + l][c0 + 2 * hf + 1];
      acc = wmma4(a, bb, acc);
    }
    const int jg  = jt * 16 + l;
    float* dst = (jg < C) ? Q : ((jg < 2 * C) ? Kf : V);
    const int col = jg & (C - 1);
#pragma unroll
    for (int r = 0; r < 8; ++r) {
      const int nrow = n0 + m0 + r + 8 * hf;
      dst[((size_t)b * N + nrow) * C + col] = acc[r];
    }
  }
}

// ---------------------------------------------------------------------------
// Kernel 2: kv partials.  kv[b,k,c,d] = sum_n Knorm[b,n,k,c]*Vrel[b,n,k,d]
// grid (SEG, KNB, B); each WG covers 512 n in 4 chunks of 128, fusing the
// position MLP, gathers, relu and l2-norm into LDS staging.
// ---------------------------------------------------------------------------
__global__ __launch_bounds__(256) void k_kv_part(
    const float* __restrict__ Kf, const float* __restrict__ V,
    const float* __restrict__ pos, const int* __restrict__ idx,
    const float* __restrict__ dist,
    const float* __restrict__ W1, const float* __restrict__ b1,
    const float* __restrict__ W2, const float* __restrict__ b2,
    float* __restrict__ kvp) {
  __shared__ float KnL[128][136];   // Knorm chunk [n][c]
  __shared__ float VrL[128][136];   // Vrel  chunk [n][d]
  __shared__ float Hl[128][66];     // MLP hidden  [n][i]
  __shared__ float W2L[128][65];    // W2 [c][i], padded vs bank conflicts
  __shared__ float W1L[64][11];
  __shared__ float b1L[64];
  __shared__ float b2L[128];
  __shared__ float SS[128];         // per-n sum of squares (ds_add_f32)
  __shared__ int   Jl[128];

  const int seg = blockIdx.x, k = blockIdx.y, b = blockIdx.z;
  const int tid = threadIdx.x;

  for (int w = tid; w < C * HID;  w += 256) W2L[w >> 6][w & 63] = W2[w];
  for (int w = tid; w < HID * 10; w += 256) W1L[w / 10][w % 10] = W1[w];
  if (tid < HID) b1L[tid] = b1[tid];
  if (tid < C)   b2L[tid] = b2[tid];

  const int wv = tid >> 5, lane = tid & 31, l = lane & 15, hf = lane >> 4;
  v8f acc[8] = {};

  for (int ch = 0; ch < 4; ++ch) {
    __syncthreads();                       // prior WMMA reads done; weights ready
    const int nb = seg * 512 + ch * 128;

    if (tid < 128) {                       // phase A: hidden layer of pos MLP
      const int n = nb + tid;
      const int j = idx[((size_t)b * N + n) * KNB + k];
      Jl[tid] = j;
      SS[tid] = 0.f;
      float f[10];
      const float pc0 = pos[((size_t)b * N + n) * 3 + 0];
      const float pc1 = pos[((size_t)b * N + n) * 3 + 1];
      const float pc2 = pos[((size_t)b * N + n) * 3 + 2];
      const float pn0 = pos[((size_t)b * N + j) * 3 + 0];
      const float pn1 = pos[((size_t)b * N + j) * 3 + 1];
      const float pn2 = pos[((size_t)b * N + j) * 3 + 2];
      f[0] = pc0; f[1] = pc1; f[2] = pc2;
      f[3] = pn0; f[4] = pn1; f[5] = pn2;
      f[6] = pc0 - pn0; f[7] = pc1 - pn1; f[8] = pc2 - pn2;
      f[9] = dist[((size_t)b * N + n) * KNB + k];
      for (int i = 0; i < HID; ++i) {
        float h = b1L[i];
#pragma unroll
        for (int q = 0; q < 10; ++q) h += f[q] * W1L[i][q];
        Hl[tid][i] = fmaxf(h, 0.f);
      }
    }
    __syncthreads();

    for (int p = 0; p < 64; ++p) {         // phase B: pos_emb + gather + sumsq
      int w  = tid + p * 256;
      int nl = w >> 7, c = w & 127;
      int j  = Jl[nl];
      float pe = b2L[c];
      for (int i = 0; i < HID; ++i) pe += W2L[c][i] * Hl[nl][i];
      float kraw = Kf[((size_t)b * N + j) * C + c] + pe;
      float vr   = fmaxf(V[((size_t)b * N + j) * C + c] + pe, 0.f);
      KnL[nl][c] = kraw;
      VrL[nl][c] = vr;
      atomicAdd(&SS[nl], kraw * kraw);
    }
    __syncthreads();

    for (int p = 0; p < 64; ++p) {         // phase C: l2 normalize K
      int w  = tid + p * 256;
      int nl = w >> 7, c = w & 127;
      KnL[nl][c] *= 1.f / fmaxf(sqrtf(SS[nl]), EPS);
    }
    __syncthreads();

    const int m0 = wv * 16;                // wave wv owns c-tile wv, all d-tiles
    for (int nn = 0; nn < 128; nn += 4) {
      v2f a;
      a.x = KnL[nn + 2 * hf][m0 + l];      // A[c][n] = Kn^T
      a.y = KnL[nn + 2 * hf + 1][m0 + l];
#pragma unroll
      for (int dt = 0; dt < 8; ++dt) {
        v2f bb;
        bb.x = VrL[nn + 2 * hf][dt * 16 + l];
        bb.y = VrL[nn + 2 * hf + 1][dt * 16 + l];
        acc[dt] = wmma4(a, bb, acc[dt]);
      }
    }
  }

  const int m0 = wv * 16;
  const size_t base = (((size_t)b * KNB + k) * SEG + seg) * (size_t)(C * C);
#pragma unroll
  for (int dt = 0; dt < 8; ++dt)
#pragma unroll
    for (int r = 0; r < 8; ++r)
      kvp[base + (size_t)(m0 + r + 8 * hf) * C + dt * 16 + l] = acc[dt][r];
}

// ---------------------------------------------------------------------------
// Kernel 3: reduce kv partials over SEG
// ---------------------------------------------------------------------------
__global__ __launch_bounds__(256) void k_kv_reduce(const float* __restrict__ kvp,
                                                   float* __restrict__ kv) {
  const size_t e  = (size_t)blockIdx.x * 256 + threadIdx.x;
  const size_t bk = e >> 14;                 // / (C*C)
  const size_t cd = e & (size_t)(C * C - 1);
  float s = 0.f;
#pragma unroll
  for (int sg = 0; sg < SEG; ++sg)
    s += kvp[(bk * SEG + sg) * (size_t)(C * C) + cd];
  kv[e] = s;
}

// ---------------------------------------------------------------------------
// Kernel 4: agg[b,n,d] = (1/N) * sum_k l2norm(Q[b,idx[b,n,k],:]) @ kv[b,k]
// kv[b,k] tile staged by TDM, overlapped with the Q gather phase.
// ---------------------------------------------------------------------------
__global__ __launch_bounds__(256) void k_agg(const float* __restrict__ Q,
                                             const int* __restrict__ idx,
                                             const float* __restrict__ kv,
                                             float* __restrict__ agg) {
  __shared__ float QnL[128][132];   // normalized gathered Q  [n][c] (A-operand)
  __shared__ float KVl[128][136];   // kv[b,k]               [c][d] (B-operand)
  __shared__ float SS[128];
  __shared__ int   Jl[128];
  const int b = blockIdx.y, n0 = blockIdx.x * 128, tid = threadIdx.x;
  const int wv = tid >> 5, lane = tid & 31, l = lane & 15, hf = lane >> 4;
  const int m0 = wv * 16;
  v8f acc[8] = {};

  for (int k = 0; k < KNB; ++k) {
    __syncthreads();                       // previous WMMA reads done
    if (tid < 128) {
      Jl[tid] = idx[((size_t)b * N + n0 + tid) * KNB + k];
      SS[tid] = 0.f;
    }
    const size_t kvb = ((size_t)b * KNB + k) * (size_t)(C * C);
#ifdef HAVE_TDM
    if (wv == 0)                           // async 64KB tile, padded to stride 136
      tdm_load_2d(&KVl[0][0], kv + kvb, 128, 128, 128ull, 6, 7, true);
#endif
    __syncthreads();
    for (int p = 0; p < 64; ++p) {
      int w  = tid + p * 256;
      int r0 = w >> 7, c = w & 127;
      float qv = Q[((size_t)b * N + Jl[r0]) * C + c];
      QnL[r0][c] = qv;
      atomicAdd(&SS[r0], qv * qv);
#ifndef HAVE_TDM
      KVl[r0][c] = kv[kvb + (size_t)r0 * C + c];
#endif
    }
#ifdef HAVE_TDM
    if (wv == 0) __builtin_amdgcn_s_wait_tensorcnt(0);
#endif
    __syncthreads();
    for (int p = 0; p < 64; ++p) {
      int w  = tid + p * 256;
      int r0 = w >> 7, c = w & 127;
      QnL[r0][c] *= 1.f / fmaxf(sqrtf(SS[r0]), EPS);
    }
    __syncthreads();
    for (int c0 = 0; c0 < C; c0 += 4) {
      v2f a;
      a.x = QnL[m0 + l][c0 + 2 * hf];
      a.y = QnL[m0 + l][c0 + 2 * hf + 1];
#pragma unroll
      for (int dt = 0; dt < 8; ++dt) {
        v2f bb;
        bb.x = KVl[c0 + 2 * hf][dt * 16 + l];
        bb.y = KVl[c0 + 2 * hf + 1][dt * 16 + l];
        acc[dt] = wmma4(a, bb, acc[dt]);
      }
    }
  }
  const float inv = 1.f / (float)N;
#pragma unroll
  for (int dt = 0; dt < 8; ++dt)
#pragma unroll
    for (int r = 0; r < 8; ++r)
      agg[((size_t)b * N + n0 + m0 + r + 8 * hf) * C + dt * 16 + l] =
          acc[dt][r] * inv;
}

// ---------------------------------------------------------------------------
// Kernel 5: out = relu(agg@Wm1^T+bm1)@Wm2^T + bm2 + xt, stored as [B,C,N]
// ---------------------------------------------------------------------------
__global__ __launch_bounds__(256) void k_mlp_out(
    const float* __restrict__ agg, const float* __restrict__ x,
    const float* __restrict__ Wm1, const float* __restrict__ bm1,
    const float* __restrict__ Wm2, const float* __restrict__ bm2,
    float* __restrict__ out) {
  __shared__ float AL[128][132];    // agg chunk [n][c]; reused for output tile
  __shared__ float HL[128][260];    // hidden half [n][h] (256 wide)
  const int b = blockIdx.y, n0 = blockIdx.x * 128, tid = threadIdx.x;
  const int wv = tid >> 5, lane = tid & 31, l = lane & 15, hf = lane >> 4;
  const int m0 = wv * 16;

#ifdef HAVE_TDM
  if (wv == 0) {                           // contiguous 64KB block, stride 132
    tdm_load_2d(&AL[0][0], agg + ((size_t)b * N + n0) * C, 128, 128,
                128ull, 6, 3, true);
    __builtin_amdgcn_s_wait_tensorcnt(0);
  }
#else
  for (int p = 0; p < 64; ++p) {
    int w = tid + p * 256;
    int nl = w >> 7, c = w & 127;
    AL[nl][c] = agg[((size_t)b * N + n0 + nl) * C + c];
  }
#endif
  v8f acc[8] = {};

  for (int half = 0; half < 2; ++half) {
    __syncthreads();                       // AL ready / HL free
    for (int ht = 0; ht < 16; ++ht) {      // H = relu(A @ Wm1h^T + bm1h)
      const int hcol = half * 256 + ht * 16 + l;
      v8f hacc = {};
      for (int c0 = 0; c0 < C; c0 += 4) {
        v2f a, bb;
        a.x  = AL[m0 + l][c0 + 2 * hf];
        a.y  = AL[m0 + l][c0 + 2 * hf + 1];
        bb.x = Wm1[(size_t)hcol * C + c0 + 2 * hf];      // L2-resident weights
        bb.y = Wm1[(size_t)hcol * C + c0 + 2 * hf + 1];
        hacc = wmma4(a, bb, hacc);
      }
      const float bias = bm1[hcol];
#pragma unroll
      for (int r = 0; r < 8; ++r)
        HL[m0 + r + 8 * hf][ht * 16 + l] = fmaxf(hacc[r] + bias, 0.f);
    }
    __syncthreads();                       // H visible
#pragma unroll
    for (int dt = 0; dt < 8; ++dt) {       // out += H @ Wm2h^T
      const int dcol = dt * 16 + l;
      for (int h0 = 0; h0 < 256; h0 += 4) {
        v2f a, bb;
        a.x  = HL[m0 + l][h0 + 2 * hf];
        a.y  = HL[m0 + l][h0 + 2 * hf + 1];
        bb.x = Wm2[(size_t)dcol * 512 + half * 256 + h0 + 2 * hf];
        bb.y = Wm2[(size_t)dcol * 512 + half * 256 + h0 + 2 * hf + 1];
        acc[dt] = wmma4(a, bb, acc[dt]);
      }
    }
  }
  __syncthreads();                         // done reading AL as input
#pragma unroll
  for (int dt = 0; dt < 8; ++dt)
#pragma unroll
    for (int r = 0; r < 8; ++r)
      AL[m0 + r + 8 * hf][dt * 16 + l] = acc[dt][r] + bm2[dt * 16 + l];
  __syncthreads();
  for (int p = 0; p < 64; ++p) {           // transpose + residual, coalesced on n
    int w = tid + p * 256;
    int d = w >> 7, nl = w & 127;
    const size_t o = ((size_t)b * C + d) * N + n0 + nl;
    out[o] = AL[nl][d] + x[o];
  }
}

// ---------------------------------------------------------------------------
extern "C" void kernel_launch(void* const* d_in, const int* in_sizes, int n_in,
                              void* d_out, int out_size, void* d_ws, size_t ws_size,
                              hipStream_t stream) {
  (void)in_sizes; (void)n_in; (void)out_size; (void)ws_size;
  const float* pos  = (const float*)d_in[0];
  const float* x    = (const float*)d_in[1];
  const int*   idx  = (const int*)d_in[2];
  const float* dist = (const float*)d_in[3];
  const float* Wqkv = (const float*)d_in[4];
  const float* W1   = (const float*)d_in[5];
  const float* b1   = (const float*)d_in[6];
  const float* W2   = (const float*)d_in[7];
  const float* b2   = (const float*)d_in[8];
  const float* Wm1  = (const float*)d_in[9];
  const float* bm1  = (const float*)d_in[10];
  const float* Wm2  = (const float*)d_in[11];
  const float* bm2  = (const float*)d_in[12];
  float* out = (float*)d_out;

  // Workspace layout (floats): Q,K,V [B,N,C] + kv partials + kv + agg (~71 MB)
  float* ws  = (float*)d_ws;
  float* Q   = ws;
  float* Kf  = Q   + (size_t)B * N * C;
  float* V   = Kf  + (size_t)B * N * C;
  float* kvp = V   + (size_t)B * N * C;
  float* kv  = kvp + (size_t)B * KNB * SEG * C * C;
  float* agg = kv  + (size_t)B * KNB * C * C;

  k_qkv<<<dim3(N / 128, B), 256, 0, stream>>>(x, Wqkv, Q, Kf, V);
  k_kv_part<<<dim3(SEG, KNB, B), 256, 0, stream>>>(Kf, V, pos, idx, dist,
                                                   W1, b1, W2, b2, kvp);
  k_kv_reduce<<<dim3((B * KNB * C * C) / 256), 256, 0, stream>>>(kvp, kv);
  k_agg<<<dim3(N / 128, B), 256, 0, stream>>>(Q, idx, kv, agg);
  k_mlp_out<<<dim3(N / 128, B), 256, 0, stream>>>(agg, x, Wm1, bm1, Wm2, bm2, out);
}